// FeatureMixPoolMixerBlock_71734543778459
// MI455X (gfx1250) — compile-verified
//
#include <hip/hip_runtime.h>
#include <math.h>

// Problem constants
#define BB 2
#define CC 64
#define SS 325
#define TT 12
#define DM 256
#define PP 32
#define IN1 768          // T*C
#define OUT2 768
#define BS 650           // B*S
#define WJ 196608        // DM*IN1 == OUT2*DM (flattened pool weight row length)
#define NXY 499200       // B*C*S*T == B*S*T*C

typedef float v2f __attribute__((ext_vector_type(2)));
typedef float v8f __attribute__((ext_vector_type(8)));

__device__ __forceinline__ float waveReduceSum(float v) {
    #pragma unroll
    for (int m = 16; m; m >>= 1) v += __shfl_xor(v, m, 32);
    return v;
}

__device__ __forceinline__ v8f wmma_f32_k4(v2f a, v2f b, v8f c) {
    return __builtin_amdgcn_wmma_f32_16x16x4_f32(
        false, a, false, b, (short)0, c, false, false);
}

// ---------------- BatchNorm statistics (channel axis = T) -----------------
__global__ __launch_bounds__(256) void bn_stats_kernel(
    const float* __restrict__ x, const float* __restrict__ gamma,
    const float* __restrict__ beta, float* __restrict__ gs, float* __restrict__ gb) {
    const int t = blockIdx.x;            // 0..11
    const int tid = threadIdx.x;
    const int lane = tid & 31, wid = tid >> 5;
    __shared__ float rs[8], rs2[8];
    const int N = BB * CC * SS;          // 41600 elems per channel
    float s = 0.f, s2 = 0.f;
    for (int i = tid; i < N; i += 256) {
        float v = x[i * TT + t];
        s += v; s2 += v * v;
    }
    s = waveReduceSum(s); s2 = waveReduceSum(s2);
    if (lane == 0) { rs[wid] = s; rs2[wid] = s2; }
    __syncthreads();
    if (tid == 0) {
        float S = 0.f, S2 = 0.f;
        for (int w = 0; w < 8; ++w) { S += rs[w]; S2 += rs2[w]; }
        float mean = S / (float)N;
        float var  = S2 / (float)N - mean * mean;
        float rstd = rsqrtf(var + 1e-5f);
        float g = gamma[t] * rstd;
        gs[t] = g;
        gb[t] = beta[t] - mean * g;
    }
}

// ---------------- y[b,s,t*64+c] = BN(x[b,c,s,t]) --------------------------
__global__ __launch_bounds__(256) void make_y_kernel(
    const float* __restrict__ x, const float* __restrict__ gs,
    const float* __restrict__ gb, float* __restrict__ y) {
    int o = blockIdx.x * 256 + threadIdx.x;
    if (o >= NXY) return;
    int bs = o / IN1, r = o % IN1;
    int t = r >> 6, c = r & 63;
    int b = bs / SS, s = bs % SS;
    y[o] = x[((b * CC + c) * SS + s) * TT + t] * gs[t] + gb[t];
}

// ---------------- key norms (32 rows x 768, 32 rows x 256) ----------------
__global__ __launch_bounds__(32) void keynorm_kernel(
    const float* __restrict__ k1, const float* __restrict__ k2,
    float* __restrict__ kn1, float* __restrict__ kn2) {
    int b = blockIdx.x, lane = threadIdx.x;
    const float* kp; int D; float* outp; int idx;
    if (b < 32) { kp = k1 + (size_t)b * IN1; D = IN1; outp = kn1; idx = b; }
    else        { kp = k2 + (size_t)(b - 32) * DM; D = DM; outp = kn2; idx = b - 32; }
    float ss = 0.f;
    for (int t = lane; t < D; t += 32) ss += kp[t] * kp[t];
    ss = waveReduceSum(ss);
    if (lane == 0) outp[idx] = sqrtf(ss);
}

// ---------------- cosine attention + softmax over P=32 --------------------
__global__ __launch_bounds__(256) void attn_kernel(
    const float* __restrict__ feat, const float* __restrict__ keys,
    const float* __restrict__ kn, float* __restrict__ aout, int D) {
    const int bs = blockIdx.x;
    const int tid = threadIdx.x, lane = tid & 31, wid = tid >> 5;
    __shared__ float red[8];
    __shared__ float simS[32];
    __shared__ float ynS;
    const float* fr = feat + (size_t)bs * D;
    // ||feat||
    float ss = 0.f;
    for (int t = tid; t < D; t += 256) { float v = fr[t]; ss += v * v; }
    ss = waveReduceSum(ss);
    if (lane == 0) red[wid] = ss;
    __syncthreads();
    if (tid == 0) {
        float tot = 0.f;
        for (int w = 0; w < 8; ++w) tot += red[w];
        ynS = sqrtf(tot);
    }
    // dot products: wave w handles p = w, w+8, w+16, w+24
    for (int p = wid; p < PP; p += 8) {
        const float* kp = keys + (size_t)p * D;
        float dp = 0.f;
        for (int t = lane; t < D; t += 32) dp += fr[t] * kp[t];
        dp = waveReduceSum(dp);
        if (lane == 0) simS[p] = dp;
    }
    __syncthreads();
    // softmax across 32 lanes of wave 0
    if (wid == 0) {
        int p = lane;
        float s = simS[p] / fmaxf(ynS * kn[p], 1e-8f);
        float m = s;
        #pragma unroll
        for (int msk = 16; msk; msk >>= 1) m = fmaxf(m, __shfl_xor(m, msk, 32));
        float e = __expf(s - m);
        float sum = e;
        #pragma unroll
        for (int msk = 16; msk; msk >>= 1) sum += __shfl_xor(sum, msk, 32);
        aout[bs * PP + p] = e / sum;
    }
}

// ---------------- pooled biases ------------------------------------------
__global__ __launch_bounds__(256) void bias_kernel(
    const float* __restrict__ a, const float* __restrict__ pb,
    float* __restrict__ outb, int D) {
    int bs = blockIdx.x;
    __shared__ float aS[PP];
    if (threadIdx.x < PP) aS[threadIdx.x] = a[bs * PP + threadIdx.x];
    __syncthreads();
    for (int d = threadIdx.x; d < D; d += 256) {
        float acc = 0.f;
        #pragma unroll 8
        for (int p = 0; p < PP; ++p) acc += aS[p] * pb[p * D + d];
        outb[(size_t)bs * D + d] = acc;
    }
}

// ---------------- WMMA pool-weight GEMM + fused contraction ---------------
// Wout[row, j] = sum_p A[row,p] * PW[p, j]     (row=bs, j in [0,196608))
// Each wave owns one "outer" index (d for pool1, tt for pool2) and sweeps the
// full rowLen reduction range, so
//   Hout[row,outer] = act(bias[row,outer] + sum_col Wtile * F[row,col])
// completes in-registers: no atomics, no re-read of the 511 MB Wout.
// FULL=true  : all 16 rows in-bounds, zero guards, 2-way N-tile unroll.
// FULL=false : tail M-tile (rows 640..655), fully guarded.
template <bool FULL>
__global__ __launch_bounds__(256) void wgen_kernel(
    const float* __restrict__ A,    // [650, 32]
    const float* __restrict__ PW,   // [32, 196608]
    const float* __restrict__ F,    // [650, rowLen]
    const float* bias,              // [650, nOuter] (aliases Hout)
    float* __restrict__ Wout,       // [650, 196608]
    float* Hout,                    // [650, nOuter] (aliases bias)
    int rowLen, int nOuter, int doRelu, int mtileBase) {
    const int tid = threadIdx.x, lane = tid & 31, wid = tid >> 5;
    const int half = lane >> 4, lq = lane & 15;
    const int outer = blockIdx.x * 8 + wid;     // d for pool1, tt for pool2
    const int bsBase = (mtileBase + (int)blockIdx.y) * 16;
    const int nTiles = rowLen >> 4;

    float rsum[8] = {0.f, 0.f, 0.f, 0.f, 0.f, 0.f, 0.f, 0.f};

    if (FULL) {
        // ---- A fragments: b64 loads, no masking ----
        const v2f* arow = (const v2f*)(A + (size_t)(bsBase + lq) * PP);
        v2f afr[8];
        #pragma unroll
        for (int kk = 0; kk < 8; ++kk) afr[kk] = arow[kk * 2 + half];

        const float* Fbase = F + (size_t)(bsBase + half * 8) * rowLen + lq;
        float* WoutBase = Wout + (size_t)(bsBase + half * 8) * WJ;
        const int jcol0 = outer * rowLen + lq;

        for (int tt = 0; tt < nTiles; tt += 2) {
            const int j0 = jcol0 + tt * 16;
            const int j1 = j0 + 16;
            __builtin_prefetch(PW + (size_t)(half * 2) * WJ + j0 + 32, 0, 1);
            v8f c0 = {}, c1 = {};
            #pragma unroll
            for (int kk = 0; kk < 8; ++kk) {
                const int p0 = kk * 4 + half * 2;
                const float* pwA = PW + (size_t)p0 * WJ;
                const float* pwB = PW + (size_t)(p0 + 1) * WJ;
                v2f b0, b1;
                b0.x = pwA[j0]; b0.y = pwB[j0];
                b1.x = pwA[j1]; b1.y = pwB[j1];
                c0 = wmma_f32_k4(afr[kk], b0, c0);
                c1 = wmma_f32_k4(afr[kk], b1, c1);
            }
            const int col0 = tt * 16;
            #pragma unroll
            for (int i = 0; i < 8; ++i) {
                const float f0 = Fbase[(size_t)i * rowLen + col0];
                const float f1 = Fbase[(size_t)i * rowLen + col0 + 16];
                rsum[i] += c0[i] * f0 + c1[i] * f1;
                WoutBase[(size_t)i * WJ + j0] = c0[i];
                WoutBase[(size_t)i * WJ + j1] = c1[i];
            }
        }
    } else {
        // ---- tail M-tile: rows 640..655, only 640..649 valid ----
        const int arow = bsBase + lq;
        const float amask = (arow < BS) ? 1.0f : 0.0f;
        const int arowc = arow < BS ? arow : (BS - 1);
        const v2f* arowp = (const v2f*)(A + (size_t)arowc * PP);
        v2f afr[8];
        #pragma unroll
        for (int kk = 0; kk < 8; ++kk) {
            afr[kk] = arowp[kk * 2 + half];
            afr[kk].x *= amask; afr[kk].y *= amask;
        }
        for (int tt = 0; tt < nTiles; ++tt) {
            const int inBase = tt * 16;
            const int j = outer * rowLen + inBase + lq;
            v8f c = {};
            #pragma unroll
            for (int kk = 0; kk < 8; ++kk) {
                const int p0 = kk * 4 + half * 2;
                v2f bfr;
                bfr.x = PW[(size_t)p0 * WJ + j];
                bfr.y = PW[(size_t)(p0 + 1) * WJ + j];
                c = wmma_f32_k4(afr[kk], bfr, c);
            }
            #pragma unroll
            for (int i = 0; i < 8; ++i) {
                const int row = bsBase + half * 8 + i;
                const int rowc = row < BS ? row : (BS - 1);
                const float fmask = (row < BS) ? 1.0f : 0.0f;
                const float fv = F[(size_t)rowc * rowLen + inBase + lq] * fmask;
                rsum[i] += c[i] * fv;
                if (row < BS) Wout[(size_t)row * WJ + j] = c[i];
            }
        }
    }

    // Reduce rsum over the 16 lanes of each half-wave (reduction columns)
    #pragma unroll
    for (int i = 0; i < 8; ++i) {
        float v = rsum[i];
        #pragma unroll
        for (int m = 8; m; m >>= 1) v += __shfl_xor(v, m, 32);
        rsum[i] = v;
    }
    if (lq == 0) {
        #pragma unroll
        for (int i = 0; i < 8; ++i) {
            const int row = bsBase + half * 8 + i;
            if (FULL || row < BS) {
                float hv = bias[(size_t)row * nOuter + outer] + rsum[i];
                if (doRelu) hv = fmaxf(hv, 0.0f);
                Hout[(size_t)row * nOuter + outer] = hv;
            }
        }
    }
}

// ---------------- residual add: out0[b,c,s,t] = x + z[b,s,t*64+c] ---------
__global__ __launch_bounds__(256) void final_kernel(
    const float* __restrict__ x, const float* __restrict__ z,
    float* __restrict__ out0) {
    int i = blockIdx.x * 256 + threadIdx.x;
    if (i >= NXY) return;
    int t = i % TT;
    int q = i / TT;
    int s = q % SS;
    int q2 = q / SS;
    int c = q2 & 63;
    int b = q2 >> 6;
    out0[i] = x[i] + z[(size_t)(b * SS + s) * OUT2 + t * CC + c];
}

extern "C" void kernel_launch(void* const* d_in, const int* in_sizes, int n_in,
                              void* d_out, int out_size, void* d_ws, size_t ws_size,
                              hipStream_t stream) {
    (void)in_sizes; (void)n_in; (void)out_size; (void)ws_size;
    const float* x        = (const float*)d_in[0];
    const float* bn_gamma = (const float*)d_in[1];
    const float* bn_beta  = (const float*)d_in[2];
    const float* keys1    = (const float*)d_in[3];
    const float* pool_w1  = (const float*)d_in[4];
    const float* pool_b1  = (const float*)d_in[5];
    const float* keys2    = (const float*)d_in[6];
    const float* pool_w2  = (const float*)d_in[7];
    const float* pool_b2  = (const float*)d_in[8];

    float* out0 = (float*)d_out;               // [B,C,S,T]
    float* w1o  = out0 + NXY;                  // [B,S,DM,IN1]
    float* w2o  = w1o + (size_t)BS * WJ;       // [B,S,OUT2,DM]

    // Workspace layout (floats)
    float* W   = (float*)d_ws;
    float* gs  = W;                            // 12 (pad 16)
    float* gb  = W + 16;                       // 12 (pad 16)
    float* y   = W + 32;                       // 650*768
    float* kn1 = y + NXY;                      // 32
    float* kn2 = kn1 + 32;                     // 32
    float* a1  = kn2 + 32;                     // 650*32
    float* hb  = a1 + BS * PP;                 // 650*256  (bias1, then h in-place)
    float* a2  = hb + BS * DM;                 // 650*32
    float* zb  = a2 + BS * PP;                 // 650*768  (bias2, then z in-place)

    bn_stats_kernel<<<TT, 256, 0, stream>>>(x, bn_gamma, bn_beta, gs, gb);
    make_y_kernel<<<(NXY + 255) / 256, 256, 0, stream>>>(x, gs, gb, y);
    keynorm_kernel<<<64, 32, 0, stream>>>(keys1, keys2, kn1, kn2);

    attn_kernel<<<BS, 256, 0, stream>>>(y, keys1, kn1, a1, IN1);
    bias_kernel<<<BS, 256, 0, stream>>>(a1, pool_b1, hb, DM);
    // w1 = A1 x PW1 (store) fused with h = relu(w1 . y + b1)
    wgen_kernel<true><<<dim3(DM / 8, 40), 256, 0, stream>>>(
        a1, pool_w1, y, hb, w1o, hb, IN1, DM, 1, 0);
    wgen_kernel<false><<<dim3(DM / 8, 1), 256, 0, stream>>>(
        a1, pool_w1, y, hb, w1o, hb, IN1, DM, 1, 40);

    attn_kernel<<<BS, 256, 0, stream>>>(hb, keys2, kn2, a2, DM);
    bias_kernel<<<BS, 256, 0, stream>>>(a2, pool_b2, zb, OUT2);
    // w2 = A2 x PW2 (store) fused with z = w2 . h + b2
    wgen_kernel<true><<<dim3(OUT2 / 8, 40), 256, 0, stream>>>(
        a2, pool_w2, hb, zb, w2o, zb, DM, OUT2, 0, 0);
    wgen_kernel<false><<<dim3(OUT2 / 8, 1), 256, 0, stream>>>(
        a2, pool_w2, hb, zb, w2o, zb, DM, OUT2, 0, 40);

    final_kernel<<<(NXY + 255) / 256, 256, 0, stream>>>(x, zb, out0);
}